// LightGCN_74320114090032
// MI455X (gfx1250) — compile-verified
//
#include <hip/hip_runtime.h>
#include <hip/hip_bf16.h>

// LightGCN K=3 propagation, CSR-gather formulation.
// Hot loop: wave32 per destination node, register accumulation, no float atomics.
// Edge stream staged into LDS with double-buffered global_load_async_to_lds_b32
// (ASYNCcnt), broadcasts via v_readlane -> scalar (saddr) row loads.

#define NN 50000
#define NE 800000
#define DF 64
#define WPB 8      // waves (nodes) per 256-thread block
#define CHUNK 32   // edges staged per async copy

// ---------------- degree histogram ----------------
__global__ void k_deg(const int* __restrict__ dst, unsigned* __restrict__ deg, int nE) {
    int e = blockIdx.x * blockDim.x + threadIdx.x;
    if (e < nE) atomicAdd(&deg[dst[e]], 1u);
}

// ---------------- norm = deg>0 ? deg^-1/2 : 1 ----------------
__global__ void k_norm(const unsigned* __restrict__ deg, float* __restrict__ norm, int n) {
    int i = blockIdx.x * blockDim.x + threadIdx.x;
    if (i < n) {
        unsigned d = deg[i];
        norm[i] = d ? rsqrtf((float)d) : 1.0f;
    }
}

// ---------------- exclusive scan of deg -> row_ptr (single workgroup) ----------------
__global__ __launch_bounds__(1024)
void k_scan(const unsigned* __restrict__ deg, int* __restrict__ row_ptr, int n) {
    __shared__ int sh[1024];
    __shared__ int run;
    const int t = threadIdx.x;
    if (t == 0) run = 0;
    __syncthreads();
    const int nchunk = (n + 1023) / 1024;
    for (int c = 0; c < nchunk; ++c) {
        int i = c * 1024 + t;
        int v = (i < n) ? (int)deg[i] : 0;
        sh[t] = v;
        __syncthreads();
        for (int off = 1; off < 1024; off <<= 1) {   // Hillis-Steele inclusive scan
            int add = (t >= off) ? sh[t - off] : 0;
            __syncthreads();
            sh[t] += add;
            __syncthreads();
        }
        int incl = sh[t];
        if (i < n) row_ptr[i] = run + (incl - v);
        __syncthreads();
        if (t == 0) run += sh[1023];
        __syncthreads();
    }
    if (t == 0) row_ptr[n] = run;   // == nE
}

// ---------------- fill CSR: per-edge weight + scatter by dst ----------------
__global__ void k_fill(const int* __restrict__ src, const int* __restrict__ dst,
                       const float* __restrict__ drop_mask, const float* __restrict__ norm,
                       const int* __restrict__ row_ptr, int* __restrict__ row_fill,
                       int* __restrict__ csr_src, float* __restrict__ csr_w, int nE) {
    int e = blockIdx.x * blockDim.x + threadIdx.x;
    if (e >= nE) return;
    int s = src[e], d = dst[e];
    float w = norm[s] * norm[d] * drop_mask[e];
    int pos = atomicAdd(&row_fill[d], 1);
    int idx = row_ptr[d] + pos;
    csr_src[idx] = s;
    csr_w[idx]   = w;
}

// ---------------- init: h0 = x, out = 0.25*x ----------------
__global__ void k_init(const float* __restrict__ x, float* __restrict__ h0,
                       float* __restrict__ out, int n) {
    int i = blockIdx.x * blockDim.x + threadIdx.x;
    if (i < n) {
        float v = x[i];
        h0[i]  = v;
        out[i] = 0.25f * v;
    }
}

// ---------------- one hop ----------------
// h_out[n] = sum_{e->n} w_e * h_in[src_e];  out += 0.25*h_out
__global__ __launch_bounds__(256)
void k_hop(const int* __restrict__ row_ptr, const int* __restrict__ csr_src,
           const float* __restrict__ csr_w, const float* __restrict__ h_in,
           float* __restrict__ h_out, float* __restrict__ out) {
    __shared__ int   sh_src[WPB][2][CHUNK];
    __shared__ float sh_w  [WPB][2][CHUNK];

    const int lane = threadIdx.x & 31;
    const int wave = threadIdx.x >> 5;
    const int node = blockIdx.x * WPB + wave;
    if (node >= NN) return;

    const int begin = row_ptr[node];
    const int end   = row_ptr[node + 1];

    // LDS byte addresses of this lane's staging slots (low 32 bits of the
    // generic pointer == LDS offset on AMDGCN)
    const unsigned lds_s[2] = { (unsigned)(size_t)&sh_src[wave][0][lane],
                                (unsigned)(size_t)&sh_src[wave][1][lane] };
    const unsigned lds_w[2] = { (unsigned)(size_t)&sh_w[wave][0][lane],
                                (unsigned)(size_t)&sh_w[wave][1][lane] };

    // async-copy 32 (src,w) pairs starting at j0 into staging buffer `buf`
    auto issue = [&](int j0, int buf) {
        int j = j0 + lane;
        if (j >= end) j = end - 1;            // clamp (duplicates are harmless)
        const int*   gs = csr_src + j;
        const float* gw = csr_w   + j;
        asm volatile("global_load_async_to_lds_b32 %0, %1, off"
                     :: "v"(lds_s[buf]), "v"((const void*)gs) : "memory");
        asm volatile("global_load_async_to_lds_b32 %0, %1, off"
                     :: "v"(lds_w[buf]), "v"((const void*)gw) : "memory");
    };

    float2 acc = make_float2(0.0f, 0.0f);
    if (begin < end) issue(begin, 0);

    int buf = 0;
    for (int j0 = begin; j0 < end; j0 += CHUNK, buf ^= 1) {
        int next = j0 + CHUNK;
        if (next < end) {
            issue(next, buf ^ 1);                              // overlap next chunk
            asm volatile("s_wait_asynccnt 0x2" ::: "memory");  // current chunk done
        } else {
            asm volatile("s_wait_asynccnt 0x0" ::: "memory");
        }

        // pull staged chunk into registers: lane k holds edge j0+k
        int   es = sh_src[wave][buf][lane];
        float ew = sh_w[wave][buf][lane];

        int cnt = end - j0; if (cnt > CHUNK) cnt = CHUNK;
        for (int k = 0; k < cnt; ++k) {
            int   s  = __builtin_amdgcn_readlane(es, k);                   // SGPR
            float wv = __uint_as_float(
                         __builtin_amdgcn_readlane(__float_as_uint(ew), k)); // SGPR
            const float2* row = (const float2*)(h_in + (size_t)s * DF);
            float2 v = row[lane];          // saddr-form global_load_b64, 256B/edge
            acc.x = fmaf(v.x, wv, acc.x);
            acc.y = fmaf(v.y, wv, acc.y);
        }
    }

    float2* orow = (float2*)(h_out + (size_t)node * DF);
    orow[lane] = acc;

    float2* arow = (float2*)(out + (size_t)node * DF);
    float2 a = arow[lane];
    a.x = fmaf(0.25f, acc.x, a.x);
    a.y = fmaf(0.25f, acc.y, a.y);
    arow[lane] = a;
}

// ---------------- host side ----------------
static inline size_t align256(size_t v) { return (v + 255) & ~(size_t)255; }

extern "C" void kernel_launch(void* const* d_in, const int* in_sizes, int n_in,
                              void* d_out, int out_size, void* d_ws, size_t ws_size,
                              hipStream_t stream) {
    const float* x         = (const float*)d_in[0];
    const int*   src       = (const int*)  d_in[1];
    const int*   dst       = (const int*)  d_in[2];
    const float* drop_mask = (const float*)d_in[3];
    float* out = (float*)d_out;

    // workspace carve-out (~33 MB total)
    char* p = (char*)d_ws;
    unsigned* deg      = (unsigned*)p;  p += align256(sizeof(unsigned) * NN);
    float*    norm     = (float*)p;     p += align256(sizeof(float) * NN);
    int*      row_ptr  = (int*)p;       p += align256(sizeof(int) * (NN + 1));
    int*      row_fill = (int*)p;       p += align256(sizeof(int) * NN);
    int*      csr_src  = (int*)p;       p += align256(sizeof(int) * NE);
    float*    csr_w    = (float*)p;     p += align256(sizeof(float) * NE);
    float*    h0       = (float*)p;     p += align256(sizeof(float) * NN * DF);
    float*    h1       = (float*)p;     p += align256(sizeof(float) * NN * DF);
    (void)ws_size; (void)n_in; (void)in_sizes; (void)out_size;

    hipMemsetAsync(deg,      0, sizeof(unsigned) * NN, stream);
    hipMemsetAsync(row_fill, 0, sizeof(int)      * NN, stream);

    const int TB = 256;
    k_deg <<<(NE + TB - 1) / TB, TB, 0, stream>>>(dst, deg, NE);
    k_norm<<<(NN + TB - 1) / TB, TB, 0, stream>>>(deg, norm, NN);
    k_scan<<<1, 1024, 0, stream>>>(deg, row_ptr, NN);
    k_fill<<<(NE + TB - 1) / TB, TB, 0, stream>>>(src, dst, drop_mask, norm,
                                                  row_ptr, row_fill, csr_src, csr_w, NE);
    k_init<<<(NN * DF + TB - 1) / TB, TB, 0, stream>>>(x, h0, out, NN * DF);

    const int hopBlocks = (NN + WPB - 1) / WPB;
    k_hop<<<hopBlocks, TB, 0, stream>>>(row_ptr, csr_src, csr_w, h0, h1, out);
    k_hop<<<hopBlocks, TB, 0, stream>>>(row_ptr, csr_src, csr_w, h1, h0, out);
    k_hop<<<hopBlocks, TB, 0, stream>>>(row_ptr, csr_src, csr_w, h0, h1, out);
}